// EnergyDecoder_7541962571953
// MI455X (gfx1250) — compile-verified
//
#include <hip/hip_runtime.h>
#include <hip/hip_bf16.h>

typedef float v2f __attribute__((ext_vector_type(2)));
typedef float v8f __attribute__((ext_vector_type(8)));

#define WMMA4(C, A, B) \
  (C) = __builtin_amdgcn_wmma_f32_16x16x4_f32(false, (A), false, (B), (short)0, (C), false, false)

__device__ __constant__ unsigned char c_wd_idx[18 * 18] = {
    0,0,0,2,0,0,0,1,1,1,1,1,4,5,6,7,9,9,
    0,1,0,2,0,0,0,1,1,1,3,1,4,5,6,7,9,9,
    0,0,1,2,0,0,0,1,1,1,1,3,4,5,6,7,9,9,
    2,2,2,4,1,0,1,2,3,3,3,3,4,5,6,7,9,9,
    0,0,0,1,0,0,0,0,0,1,1,1,4,5,6,6,7,8,
    0,0,0,0,0,1,1,2,2,3,3,3,4,5,6,6,7,8,
    0,0,0,1,0,1,1,2,2,2,2,2,4,4,5,5,6,7,
    1,1,1,2,0,2,2,2,2,2,2,2,3,4,4,5,6,7,
    1,1,1,3,0,2,2,2,2,2,2,2,3,4,4,4,5,6,
    1,1,1,3,1,3,2,2,2,2,2,2,4,4,5,6,6,6,
    1,3,1,3,1,3,2,2,2,2,3,2,4,4,5,6,6,6,
    1,1,3,3,1,3,2,2,2,2,2,3,4,4,5,6,6,6,
    4,4,4,4,4,4,4,3,3,4,4,4,4,4,4,4,5,6,
    5,5,5,5,5,5,4,4,4,4,4,4,4,4,4,4,5,5,
    6,6,6,6,6,6,5,4,4,5,5,5,4,4,4,4,5,5,
    7,7,7,7,6,6,5,5,4,6,6,6,4,4,4,4,4,4,
    9,9,9,9,7,7,6,6,5,6,6,6,5,5,5,4,4,4,
    9,9,9,9,8,8,7,7,6,6,6,6,6,5,5,4,4,4};

__device__ __constant__ float c_levels[10] = {5.6f, 5.0f, 4.5f, 4.0f, 3.5f,
                                              3.1f, 2.7f, 2.3f, 2.0f, 2.0f};

#define WAVES_PER_BLOCK 8
#define BLOCK_THREADS 256
#define PAIRS_PER_CHUNK 256
#define MAXG 64

__device__ __forceinline__ float softplusf(float x) {
  return fmaxf(x, 0.f) + log1pf(expf(-fabsf(x)));
}

__global__ void energy_zero_kernel(double* bins, int n) {
  int t = threadIdx.x + blockIdx.x * blockDim.x;
  if (t < n) bins[t] = 0.0;
}

__global__ __launch_bounds__(BLOCK_THREADS) void energy_pair_kernel(
    const float* __restrict__ node_emb, const float* __restrict__ positions,
    const float* __restrict__ wA, const float* __restrict__ bA,
    const float* __restrict__ wB, const float* __restrict__ bB,
    const int* __restrict__ pairs, const int* __restrict__ ptype,
    const int* __restrict__ node2graph, double* __restrict__ bins, int P, int G) {
  __shared__ float s_w[3][128];                    // wA, wB, zeros (for unused B cols)
  __shared__ float s_well[18 * 18];                // CG_WELL staged
  __shared__ float s_dot[WAVES_PER_BLOCK][64];     // [col*32 + pairInWave]
  __shared__ double s_bins[4 * MAXG];

  const int tid = threadIdx.x;
  if (tid < 128) {
    s_w[0][tid] = wA[tid];
    s_w[1][tid] = wB[tid];
    s_w[2][tid] = 0.f;
  }
  for (int i = tid; i < 18 * 18; i += BLOCK_THREADS)
    s_well[i] = c_levels[c_wd_idx[i]];
  for (int i = tid; i < 4 * G; i += BLOCK_THREADS) s_bins[i] = 0.0;
  __syncthreads();

  const int wid = tid >> 5;
  const int lane = tid & 31;
  const int n = lane & 15;       // A-row / B-column index within tile
  const int hi = lane >> 4;      // 0: K=0,1  1: K=2,3
  const int koff = hi * 2;
  const float* wrow = s_w[(n < 2) ? n : 2];
  const float bA0 = bA[0];
  const float bB0 = bB[0];

  const int nChunks = (P + PAIRS_PER_CHUNK - 1) / PAIRS_PER_CHUNK;
  for (int chunk = blockIdx.x; chunk < nChunks; chunk += gridDim.x) {
    const int waveBase = chunk * PAIRS_PER_CHUNK + wid * 32;

    // ---- WMMA phase: two 16-pair tiles, D = pe(16x128) @ [wA wB 0...](128x16)
    for (int t = 0; t < 2; ++t) {
      int pIdx = waveBase + t * 16 + n;
      int p = pIdx < P ? pIdx : P - 1;
      const int ia = pairs[2 * p];
      const int ja = pairs[2 * p + 1];
      const float* rowI = node_emb + (long)ia * 64;
      const float* rowJ = node_emb + (long)ja * 64;
      v8f c0 = {}; v8f c1 = {};
#pragma unroll
      for (int kk = 0; kk < 16; ++kk) {         // K = 0..63 -> emb[i]
        int kabs = kk * 4 + koff;
        v2f a = *(const v2f*)(rowI + kabs);
        v2f b; b.x = wrow[kabs]; b.y = wrow[kabs + 1];
        if (kk & 1) { WMMA4(c1, a, b); } else { WMMA4(c0, a, b); }
      }
#pragma unroll
      for (int kk = 16; kk < 32; ++kk) {        // K = 64..127 -> emb[j]
        int kabs = kk * 4 + koff;
        v2f a = *(const v2f*)(rowJ + (kabs - 64));
        v2f b; b.x = wrow[kabs]; b.y = wrow[kabs + 1];
        if (kk & 1) { WMMA4(c1, a, b); } else { WMMA4(c0, a, b); }
      }
      v8f c = c0 + c1;
      if (n < 2) {  // lanes 0,1,16,17 carry D columns 0 (dotA) and 1 (dotB)
        float* dst = &s_dot[wid][n * 32 + t * 16 + hi * 8];
#pragma unroll
        for (int r = 0; r < 8; ++r) dst[r] = c[r];
      }
    }
    __syncthreads();

    // ---- scalar phase: lane L owns pair waveBase+L
    {
      int pIdx = waveBase + lane;
      bool valid = pIdx < P;
      int p = valid ? pIdx : P - 1;
      float dotA = s_dot[wid][lane] + bA0;
      float dotB = s_dot[wid][32 + lane] + bB0;

      int t0 = ptype[2 * p];
      int t1 = ptype[2 * p + 1];
      int ia = pairs[2 * p];
      int ja = pairs[2 * p + 1];
      float dx = positions[3 * ia] - positions[3 * ja];
      float dy = positions[3 * ia + 1] - positions[3 * ja + 1];
      float dz = positions[3 * ia + 2] - positions[3 * ja + 2];
      float dm = sqrtf(dx * dx + dy * dy + dz * dz);

      bool ring = (t0 >= 13) & (t0 <= 16) & (t1 >= 13) & (t1 <= 16);
      float Bv = tanhf(dotB) * 0.2f;
      float vdw = (((t0 < 4) & (t1 >= 16)) | ((t0 >= 16) & (t1 < 4))) ? 6.2f : 4.7f;
      float dm0 = (ring ? 4.3f : vdw) + Bv;
      float ratio = dm0 / dm;
      float r2 = ratio * ratio;
      float r6 = r2 * r2 * r2;
      float x = r6 * r6 - r6;
      float e = (x > 100.f) ? 100.f : x;  // NaN propagates like jnp.minimum
      float A_ = s_well[t0 * 18 + t1] * (ring ? 0.75f : 1.f);
      float A = fminf(fmaxf(softplusf(dotA) * A_, 1.5f), 5.6f);
      float LJ = A * e;
      bool ch = ((t0 == 1) | (t0 == 2)) & ((t1 == 1) | (t1 == 2));
      float q0 = (t0 == 1) ? 1.f : ((t0 == 2) ? -1.f : 0.f);
      float q1 = (t1 == 1) ? 1.f : ((t1 == 2) ? -1.f : 0.f);
      float ES = ch ? (q0 * q1) / dm : 0.f;
      int g = node2graph[ia];

      if (valid) {
        atomicAdd(&s_bins[g], 1.0);
        atomicAdd(&s_bins[G + g], (double)LJ);
        if (ch) {
          atomicAdd(&s_bins[2 * G + g], 1.0);
          atomicAdd(&s_bins[3 * G + g], (double)ES);
        }
      }
    }
    __syncthreads();  // protect s_dot WAR for next chunk
  }

  __syncthreads();
  for (int i = tid; i < 4 * G; i += BLOCK_THREADS) atomicAdd(&bins[i], s_bins[i]);
}

__global__ void energy_final_kernel(const double* __restrict__ bins,
                                    const float* __restrict__ graph_emb,
                                    const float* __restrict__ w_lj,
                                    const float* __restrict__ b_lj,
                                    const float* __restrict__ w_es,
                                    const float* __restrict__ b_es,
                                    const float* __restrict__ w_mlp,
                                    const float* __restrict__ b_mlp,
                                    float* __restrict__ out, int G, int D) {
  int g = threadIdx.x;
  if (g >= G) return;
  float cnt = (float)bins[g];
  float ljs = (float)bins[G + g];
  float ccnt = (float)bins[2 * G + g];
  float ess = (float)bins[3 * G + g];
  float lj_mean = ljs / fmaxf(cnt, 1.f);
  float es_mean = ess / fmaxf(ccnt, 1.f);
  float LJp = lj_mean * w_lj[0] + b_lj[0];
  float xe = es_mean * w_es[0] + b_es[0];
  float ESp = fmaxf(xe, 0.f) + log1pf(expf(-fabsf(xe)));
  float fe = LJp + ESp;
  float s = 0.f, wsum = 0.f;
  for (int d = 0; d < D; ++d) {
    float w = w_mlp[d];
    s += graph_emb[g * D + d] * w;
    wsum += w;
  }
  out[g] = s + fe * wsum + b_mlp[0];
}

extern "C" void kernel_launch(void* const* d_in, const int* in_sizes, int n_in,
                              void* d_out, int out_size, void* d_ws, size_t ws_size,
                              hipStream_t stream) {
  const float* node_emb = (const float*)d_in[0];
  const float* positions = (const float*)d_in[1];
  const float* graph_emb = (const float*)d_in[2];
  const float* wA = (const float*)d_in[3];
  const float* bA = (const float*)d_in[4];
  const float* wB = (const float*)d_in[5];
  const float* bB = (const float*)d_in[6];
  const float* w_lj = (const float*)d_in[7];
  const float* b_lj = (const float*)d_in[8];
  const float* w_es = (const float*)d_in[9];
  const float* b_es = (const float*)d_in[10];
  const float* w_mlp = (const float*)d_in[11];
  const float* b_mlp = (const float*)d_in[12];
  const int* pairs = (const int*)d_in[13];
  const int* ptype = (const int*)d_in[14];
  const int* n2g = (const int*)d_in[15];

  const int N = in_sizes[1] / 3;
  const int D = in_sizes[0] / N;        // 64
  const int G = in_sizes[2] / D;        // 16
  const int P = in_sizes[13] / 2;       // 1e6
  (void)N; (void)n_in; (void)ws_size; (void)out_size;

  double* bins = (double*)d_ws;         // 4*G doubles

  energy_zero_kernel<<<1, 256, 0, stream>>>(bins, 4 * G);

  const int nChunks = (P + PAIRS_PER_CHUNK - 1) / PAIRS_PER_CHUNK;
  int blocks = nChunks < 512 ? nChunks : 512;
  energy_pair_kernel<<<blocks, BLOCK_THREADS, 0, stream>>>(
      node_emb, positions, wA, bA, wB, bB, pairs, ptype, n2g, bins, P, G);

  energy_final_kernel<<<1, 64, 0, stream>>>(bins, graph_emb, w_lj, b_lj, w_es,
                                            b_es, w_mlp, b_mlp, (float*)d_out, G, D);
}